// TopologicalSignatureDistanceWC_33234456937220
// MI455X (gfx1250) — compile-verified
//
#include <hip/hip_runtime.h>

#define N     2048
#define DLAT  32
#define KNN   20
#define NT    128                 // 16x16 tiles per dimension
#define NTILES (NT * NT)          // 16384 tiles
#define NRBLK 4096                // reduction blocks

typedef __attribute__((ext_vector_type(2))) float v2f;
typedef __attribute__((ext_vector_type(8))) float v8f;

// workspace byte offsets
#define WS_SQ    ((size_t)0)
#define WS_DIST  ((size_t)8192)
#define WS_MASK  (WS_DIST + (size_t)N * N * 4)
#define WS_PART  (WS_MASK + (size_t)N * N)

// ---------------------------------------------------------------------------
// Kernel 1: per-row sum of squares of latent  (2048 rows x 32)
// ---------------------------------------------------------------------------
__global__ void k_rowsq(const float* __restrict__ latent, float* __restrict__ sq) {
    int row = blockIdx.x * blockDim.x + threadIdx.x;
    if (row < N) {
        const float4* p = (const float4*)(latent + (size_t)row * DLAT);
        float s = 0.0f;
#pragma unroll
        for (int i = 0; i < DLAT / 4; ++i) {
            float4 v = p[i];
            s += v.x * v.x + v.y * v.y + v.z * v.z + v.w * v.w;
        }
        sq[row] = s;
    }
}

// ---------------------------------------------------------------------------
// Kernel 2: dist_Z via WMMA f32 16x16x4 Gram matmul.
// One wave32 computes one 16x16 tile of G = latent @ latent^T, then
// dist_Z = sqrt(max(sq_i + sq_j - 2G, 0)) / latent_norm.
// A layout (32-bit 16x4): lanes 0-15 -> K={k0,k0+1}, lanes 16-31 -> K={k0+2,k0+3},
// M = lane%16. B mirrors A (B = A^T for the Gram).
// C/D layout: element (M = v + 8*(lane/16), N = lane%16) in VGPR v.
// ---------------------------------------------------------------------------
__global__ void k_dist(const float* __restrict__ latent,
                       const float* __restrict__ sq,
                       const float* __restrict__ lat_norm,
                       float* __restrict__ distZ) {
    const int lane = threadIdx.x & 31;
    const int wave = threadIdx.x >> 5;
    const int tile = blockIdx.x * 8 + wave;
    const int ti   = tile >> 7;      // tile / 128
    const int tj   = tile & 127;     // tile % 128
    const int half = lane >> 4;
    const int sub  = lane & 15;

    const int arow = ti * 16 + sub;  // row of A this lane supplies
    const int brow = tj * 16 + sub;  // row of latent acting as B column

    v8f c = {};
#pragma unroll
    for (int k0 = 0; k0 < DLAT; k0 += 4) {
        int k = k0 + 2 * half;
        v2f a = *(const v2f*)(latent + (size_t)arow * DLAT + k);
        v2f b = *(const v2f*)(latent + (size_t)brow * DLAT + k);
        c = __builtin_amdgcn_wmma_f32_16x16x4_f32(
                /*neg_a=*/false, a, /*neg_b=*/false, b,
                /*c_mod=*/(short)0, c, /*reuse_a=*/false, /*reuse_b=*/false);
    }

    const float inv  = 1.0f / lat_norm[0];
    const int   nidx = tj * 16 + sub;
    const float sqn  = sq[nidx];
#pragma unroll
    for (int v = 0; v < 8; ++v) {
        int   m  = ti * 16 + v + 8 * half;
        float d2 = sq[m] + sqn - 2.0f * c[v];
        d2 = fmaxf(d2, 0.0f);
        float d = (d2 > 0.0f) ? (__builtin_sqrtf(d2) * inv) : 0.0f;
        distZ[(size_t)m * N + nidx] = d;
    }
}

// ---------------------------------------------------------------------------
// Kernel 3: row-wise kNN mask. One 256-thread block per row.
// Key = (float_bits << 32) | col  -> ascending key order == stable argsort
// (value asc, index asc) for non-negative floats. 21 iterative mins; skip
// rank 0 (self), mark ranks 1..20.
// ---------------------------------------------------------------------------
__global__ void k_knn(const float* __restrict__ distZ, unsigned char* __restrict__ maskZ) {
    __shared__ unsigned long long red[256];
    const int row = blockIdx.x;
    const int tid = threadIdx.x;

    // zero this row of maskZ (2048 bytes, 8 per thread via u64)
    {
        unsigned long long* mz = (unsigned long long*)(maskZ + (size_t)row * N);
        mz[tid] = 0ull;
    }
    __syncthreads();

    unsigned long long key[8];
#pragma unroll
    for (int i = 0; i < 8; ++i) {
        int j = tid + i * 256;
        unsigned int bits = __float_as_uint(distZ[(size_t)row * N + j]);
        key[i] = ((unsigned long long)bits << 32) | (unsigned int)j;
    }

    unsigned long long prev = 0ull;
    for (int it = 0; it <= KNN; ++it) {
        unsigned long long lm = ~0ull;
#pragma unroll
        for (int i = 0; i < 8; ++i) {
            unsigned long long k = key[i];
            if ((it == 0 || k > prev) && k < lm) lm = k;
        }
        red[tid] = lm;
        __syncthreads();
        for (int s = 128; s > 0; s >>= 1) {
            if (tid < s) {
                unsigned long long o = red[tid + s];
                if (o < red[tid]) red[tid] = o;
            }
            __syncthreads();
        }
        unsigned long long mk = red[0];
        __syncthreads();   // red[] reused next iteration
        if (it > 0) {
            int j = (int)(mk & 0xffffffffull);
            if ((j & 255) == tid)           // unique owner writes
                maskZ[(size_t)row * N + j] = 1;
        }
        prev = mk;
    }
}

// ---------------------------------------------------------------------------
// Kernel 4: per-block partial sums of the four reductions (deterministic).
//   p0 = sum maskX * (dX - dZ)^2        (distance1_2)
//   p1 = sum maskZ * (dX - dZ)^2        (distance2_1)
//   p2 = sum (mZ | mZ^T)                (mZtot total)
//   p3 = sum ((mZ|mZ^T) & !(mX|mX^T))   (non-matching count)
// ---------------------------------------------------------------------------
__global__ void k_reduce(const float* __restrict__ distX,
                         const float* __restrict__ maskX,
                         const float* __restrict__ distZ,
                         const unsigned char* __restrict__ maskZ,
                         float* __restrict__ part) {
    __shared__ float s0[256], s1[256], s2[256], s3[256];
    const int tid = threadIdx.x;
    float a0 = 0.f, a1 = 0.f, a2 = 0.f, a3 = 0.f;
#pragma unroll
    for (int k = 0; k < 4; ++k) {
        size_t e = (size_t)blockIdx.x * 1024 + (size_t)k * 256 + tid;
        int i = (int)(e >> 11);
        int j = (int)(e & 2047);
        float dx   = distX[e];
        float dz   = distZ[e];
        float mx   = maskX[e];
        float diff = dx - dz;
        float sq   = diff * diff;
        float mzf  = (float)maskZ[e];
        a0 += mx * sq;
        a1 += mzf * sq;
        bool mzt = (maskZ[e] != 0) || (maskZ[(size_t)j * N + i] != 0);
        bool mxt = (mx != 0.0f) || (maskX[(size_t)j * N + i] != 0.0f);
        a2 += mzt ? 1.0f : 0.0f;
        a3 += (mzt && !mxt) ? 1.0f : 0.0f;
    }
    s0[tid] = a0; s1[tid] = a1; s2[tid] = a2; s3[tid] = a3;
    __syncthreads();
    for (int s = 128; s > 0; s >>= 1) {
        if (tid < s) {
            s0[tid] += s0[tid + s];
            s1[tid] += s1[tid + s];
            s2[tid] += s2[tid + s];
            s3[tid] += s3[tid + s];
        }
        __syncthreads();
    }
    if (tid == 0) {
        part[blockIdx.x * 4 + 0] = s0[0];
        part[blockIdx.x * 4 + 1] = s1[0];
        part[blockIdx.x * 4 + 2] = s2[0];
        part[blockIdx.x * 4 + 3] = s3[0];
    }
}

// ---------------------------------------------------------------------------
// Kernel 5: combine partials in a fixed order; emit the 4 outputs.
// ---------------------------------------------------------------------------
__global__ void k_final(const float* __restrict__ part, float* __restrict__ out) {
    __shared__ float sh[4][256];
    const int tid = threadIdx.x;
    float a[4] = {0.f, 0.f, 0.f, 0.f};
    for (int b = tid; b < NRBLK; b += 256)
#pragma unroll
        for (int q = 0; q < 4; ++q) a[q] += part[b * 4 + q];
#pragma unroll
    for (int q = 0; q < 4; ++q) sh[q][tid] = a[q];
    __syncthreads();
    for (int s = 128; s > 0; s >>= 1) {
        if (tid < s)
#pragma unroll
            for (int q = 0; q < 4; ++q) sh[q][tid] += sh[q][tid + s];
        __syncthreads();
    }
    if (tid == 0) {
        float d1 = sh[0][0], d2 = sh[1][0], mzt = sh[2][0], nm = sh[3][0];
        out[0] = d1 + d2;   // distance
        out[1] = nm / mzt;  // non_matching_pairs
        out[2] = d1;        // distance1_2
        out[3] = d2;        // distance2_1
    }
}

// ---------------------------------------------------------------------------
extern "C" void kernel_launch(void* const* d_in, const int* in_sizes, int n_in,
                              void* d_out, int out_size, void* d_ws, size_t ws_size,
                              hipStream_t stream) {
    const float* latent   = (const float*)d_in[0];   // [2048,32]
    const float* lat_norm = (const float*)d_in[1];   // [1]
    const float* distX    = (const float*)d_in[2];   // [2048,2048]
    const float* maskX    = (const float*)d_in[3];   // [2048,2048]
    float*       out      = (float*)d_out;           // 4 scalars

    char* ws = (char*)d_ws;
    float*         sq    = (float*)(ws + WS_SQ);
    float*         distZ = (float*)(ws + WS_DIST);
    unsigned char* maskZ = (unsigned char*)(ws + WS_MASK);
    float*         part  = (float*)(ws + WS_PART);

    k_rowsq <<<N / 256, 256, 0, stream>>>(latent, sq);
    k_dist  <<<NTILES / 8, 256, 0, stream>>>(latent, sq, lat_norm, distZ);
    k_knn   <<<N, 256, 0, stream>>>(distZ, maskZ);
    k_reduce<<<NRBLK, 256, 0, stream>>>(distX, maskX, distZ, maskZ, part);
    k_final <<<1, 256, 0, stream>>>(part, out);
}